// Net_41506563949058
// MI455X (gfx1250) — compile-verified
//
#include <hip/hip_runtime.h>
#include <math.h>

#define N_NODES   100000
#define N_EDGES   3200000
#define IN_FEATS  512
#define HIDDEN    16
#define NUM_CLASSES 40
#define NPAD      48              // NUM_CLASSES padded to 3 WMMA N-tiles
#define M_TILES   (N_NODES / 16)  // 6250 exactly
#define WAVES_PER_BLOCK 8
#define TPB 256
#define KC 64                     // K-chunk (floats) staged in LDS per step
#define CHUNKS (IN_FEATS / KC)    // 8

typedef __attribute__((ext_vector_type(2))) float v2f;
typedef __attribute__((ext_vector_type(8))) float v8f;

// ---------------- degree / norm ----------------

__global__ void init_deg_kernel(unsigned int* deg) {
    int n = blockIdx.x * blockDim.x + threadIdx.x;
    if (n < N_NODES) deg[n] = 1u;  // self-loop
}

__global__ void prep_edges_kernel(const long long* __restrict__ src64,
                                  const long long* __restrict__ dst64,
                                  int* __restrict__ src32,
                                  int* __restrict__ dst32,
                                  unsigned int* __restrict__ deg) {
    int e = blockIdx.x * blockDim.x + threadIdx.x;
    if (e >= N_EDGES) return;
    int s = (int)src64[e];
    int d = (int)dst64[e];
    src32[e] = s;
    dst32[e] = d;
    atomicAdd(&deg[d], 1u);
}

__global__ void dinv_kernel(const unsigned int* __restrict__ deg,
                            float* __restrict__ dinv) {
    int n = blockIdx.x * blockDim.x + threadIdx.x;
    if (n >= N_NODES) return;
    float d = (float)deg[n];
    dinv[n] = (d > 0.0f) ? rsqrtf(d) : 0.0f;
}

// ---------------- layer 1: h1 = x @ W1 (WMMA fp32 16x16x4) ----------------
// x tile (16 rows x KC cols) is staged to LDS with async global->LDS copies
// (ASYNCcnt), double-buffered per wave: chunk c+1 streams in while chunk c
// feeds the WMMA pipe. Staging loads are fully coalesced b128s.

__global__ void __launch_bounds__(TPB)
gemm1_kernel(const float* __restrict__ x,
             const float* __restrict__ W1,
             float* __restrict__ h1) {
    __shared__ float w1s[IN_FEATS * HIDDEN];              // 32 KB
    __shared__ float xs[WAVES_PER_BLOCK * 2 * 16 * KC];   // 64 KB
    for (int i = threadIdx.x; i < IN_FEATS * HIDDEN; i += blockDim.x)
        w1s[i] = W1[i];
    __syncthreads();

    const int wave = threadIdx.x >> 5;
    const int lane = threadIdx.x & 31;
    const int tile = blockIdx.x * WAVES_PER_BLOCK + wave;
    if (tile >= M_TILES) return;          // wave-uniform: EXEC all-1 inside

    const int half = lane >> 4;           // K-pair selector
    const int n    = lane & 15;           // A row (M) == B col (N) per lane
    const int m0   = tile * 16;

    float* myxs = xs + wave * (2 * 16 * KC);

    // async-stage one 16 x KC chunk of x into LDS buffer `buf`
    auto issue_chunk = [&](int buf, int c) {
        const float* gbase = x + (size_t)m0 * IN_FEATS + c * KC;
        float* lbase = myxs + buf * (16 * KC);
        #pragma unroll
        for (int i = 0; i < 8; ++i) {                 // 8 * 32 lanes * 16B = 4KB
            int flat = i * 32 + lane;                 // 0..255 transfers
            int row  = flat >> 4;                     // 16 xfers per row
            int col4 = (flat & 15) << 2;              // float column
            unsigned loff = (unsigned)(uintptr_t)(lbase + row * KC + col4);
            unsigned long long gaddr =
                (unsigned long long)(uintptr_t)(gbase + (size_t)row * IN_FEATS + col4);
            asm volatile("global_load_async_to_lds_b128 %0, %1, off"
                         :: "v"(loff), "v"(gaddr) : "memory");
        }
    };

    v8f cacc = {0.f, 0.f, 0.f, 0.f, 0.f, 0.f, 0.f, 0.f};
    issue_chunk(0, 0);
    for (int c = 0; c < CHUNKS; ++c) {
        if (c + 1 < CHUNKS) {
            issue_chunk((c + 1) & 1, c + 1);          // prefetch next chunk
            asm volatile("s_wait_asynccnt 8" ::: "memory");  // chunk c done
        } else {
            asm volatile("s_wait_asynccnt 0" ::: "memory");
        }
        const float* abuf  = myxs + (c & 1) * (16 * KC) + n * KC;
        const float* bbase = w1s + (c * KC) * HIDDEN;
        #pragma unroll 4
        for (int kk = 0; kk < KC; kk += 4) {
            const int kb = kk + 2 * half;
            v2f a = *(const v2f*)(abuf + kb);         // A from LDS (ds b64)
            v2f b;
            b.x = bbase[kb * HIDDEN + n];             // B: W1[k][n] from LDS
            b.y = bbase[(kb + 1) * HIDDEN + n];
            cacc = __builtin_amdgcn_wmma_f32_16x16x4_f32(
                    false, a, false, b, (short)0, cacc, false, false);
        }
    }

    #pragma unroll
    for (int r = 0; r < 8; ++r)
        h1[(size_t)(m0 + r + 8 * half) * HIDDEN + n] = cacc[r];
}

// agg1 = b1 + dinv^2 * h1 (self-loop term)
__global__ void init_agg1_kernel(const float* __restrict__ h1,
                                 const float* __restrict__ b1,
                                 const float* __restrict__ dinv,
                                 float* __restrict__ agg1) {
    int idx = blockIdx.x * blockDim.x + threadIdx.x;
    if (idx >= N_NODES * HIDDEN) return;
    int node = idx >> 4;
    int f    = idx & 15;
    float dv = dinv[node];
    agg1[idx] = b1[f] + dv * dv * h1[idx];
}

__global__ void scatter1_kernel(const float* __restrict__ h1,
                                const int* __restrict__ src32,
                                const int* __restrict__ dst32,
                                const float* __restrict__ dinv,
                                float* __restrict__ agg1) {
    int t = blockIdx.x * blockDim.x + threadIdx.x;
    if (t >= N_EDGES * HIDDEN) return;
    int e = t >> 4;
    int f = t & 15;
    int s = src32[e];
    int d = dst32[e];
    float w = dinv[s] * dinv[d];
    atomicAdd(&agg1[d * HIDDEN + f], h1[s * HIDDEN + f] * w);
}

// ---------------- layer 2: z2 = relu(agg1) @ W2 (WMMA, N padded to 48) ----

__global__ void __launch_bounds__(TPB)
gemm2_kernel(const float* __restrict__ agg1,
             const float* __restrict__ W2,
             float* __restrict__ z2) {
    __shared__ float w2s[HIDDEN * NPAD];  // 3 KB, zero-padded cols 40..47
    for (int i = threadIdx.x; i < HIDDEN * NPAD; i += blockDim.x) {
        int k = i / NPAD, nn = i % NPAD;
        w2s[i] = (nn < NUM_CLASSES) ? W2[k * NUM_CLASSES + nn] : 0.0f;
    }
    __syncthreads();

    const int wave = threadIdx.x >> 5;
    const int lane = threadIdx.x & 31;
    const int tile = blockIdx.x * WAVES_PER_BLOCK + wave;
    if (tile >= M_TILES) return;          // wave-uniform

    const int half = lane >> 4;
    const int n    = lane & 15;
    const float* arow = agg1 + (size_t)(tile * 16 + n) * HIDDEN;

    // Load the lane's 8 A values once (K=16), fuse ReLU.
    v2f a[4];
    #pragma unroll
    for (int s = 0; s < 4; ++s) {
        v2f t = *(const v2f*)(arow + 4 * s + 2 * half);
        t.x = fmaxf(t.x, 0.0f);
        t.y = fmaxf(t.y, 0.0f);
        a[s] = t;
    }

    const int m0 = tile * 16;
    #pragma unroll
    for (int nt = 0; nt < 3; ++nt) {
        v8f c = {0.f, 0.f, 0.f, 0.f, 0.f, 0.f, 0.f, 0.f};
        #pragma unroll
        for (int s = 0; s < 4; ++s) {
            const int kb = 4 * s + 2 * half;
            v2f b;
            b.x = w2s[kb * NPAD + nt * 16 + n];
            b.y = w2s[(kb + 1) * NPAD + nt * 16 + n];
            c = __builtin_amdgcn_wmma_f32_16x16x4_f32(
                    false, a[s], false, b, (short)0, c, false, false);
        }
        const int col = nt * 16 + n;
        if (col < NUM_CLASSES) {          // divergent only at store, after WMMA
            #pragma unroll
            for (int r = 0; r < 8; ++r)
                z2[(size_t)(m0 + r + 8 * half) * NUM_CLASSES + col] = c[r];
        }
    }
}

// out = b2 + dinv^2 * z2 (self-loop term), accumulates edges on top
__global__ void init_out_kernel(const float* __restrict__ z2,
                                const float* __restrict__ b2,
                                const float* __restrict__ dinv,
                                float* __restrict__ out) {
    int idx = blockIdx.x * blockDim.x + threadIdx.x;
    if (idx >= N_NODES * NUM_CLASSES) return;
    int node = idx / NUM_CLASSES;
    int f    = idx - node * NUM_CLASSES;
    float dv = dinv[node];
    out[idx] = b2[f] + dv * dv * z2[idx];
}

__global__ void scatter2_kernel(const float* __restrict__ z2,
                                const int* __restrict__ src32,
                                const int* __restrict__ dst32,
                                const float* __restrict__ dinv,
                                float* __restrict__ out) {
    int t = blockIdx.x * blockDim.x + threadIdx.x;
    if (t >= N_EDGES * NUM_CLASSES) return;
    int e = t / NUM_CLASSES;
    int f = t - e * NUM_CLASSES;
    int s = src32[e];
    int d = dst32[e];
    float w = dinv[s] * dinv[d];
    atomicAdd(&out[d * NUM_CLASSES + f], z2[s * NUM_CLASSES + f] * w);
}

// ---------------- log_softmax (in place, one thread per node) -------------

__global__ void log_softmax_kernel(float* __restrict__ out) {
    int node = blockIdx.x * blockDim.x + threadIdx.x;
    if (node >= N_NODES) return;
    float* row = out + (size_t)node * NUM_CLASSES;
    float v[NUM_CLASSES];
    float m = -INFINITY;
    #pragma unroll
    for (int f = 0; f < NUM_CLASSES; ++f) {
        v[f] = row[f];
        m = fmaxf(m, v[f]);
    }
    float se = 0.0f;
    #pragma unroll
    for (int f = 0; f < NUM_CLASSES; ++f)
        se += __expf(v[f] - m);
    float lse = m + __logf(se);
    #pragma unroll
    for (int f = 0; f < NUM_CLASSES; ++f)
        row[f] = v[f] - lse;
}

// ---------------- launch ---------------------------------------------------

extern "C" void kernel_launch(void* const* d_in, const int* in_sizes, int n_in,
                              void* d_out, int out_size, void* d_ws, size_t ws_size,
                              hipStream_t stream) {
    const float*     x    = (const float*)d_in[0];
    const long long* ei   = (const long long*)d_in[1];   // int64 [2, E]
    const float*     W1   = (const float*)d_in[2];
    const float*     b1   = (const float*)d_in[3];
    const float*     W2   = (const float*)d_in[4];
    const float*     b2   = (const float*)d_in[5];
    float*           out  = (float*)d_out;

    const long long* src64 = ei;
    const long long* dst64 = ei + N_EDGES;

    // workspace carve-up (256B aligned)
    char* ws = (char*)d_ws;
    size_t off = 0;
    auto carve = [&](size_t bytes) {
        void* p = ws + off;
        off = (off + bytes + 255) & ~(size_t)255;
        return p;
    };
    int*          src32 = (int*)carve((size_t)N_EDGES * 4);
    int*          dst32 = (int*)carve((size_t)N_EDGES * 4);
    unsigned int* deg   = (unsigned int*)carve((size_t)N_NODES * 4);
    float*        dinv  = (float*)carve((size_t)N_NODES * 4);
    float*        h1    = (float*)carve((size_t)N_NODES * HIDDEN * 4);
    float*        agg1  = (float*)carve((size_t)N_NODES * HIDDEN * 4);
    float*        z2    = (float*)carve((size_t)N_NODES * NUM_CLASSES * 4);

    const int gemm_grid = (M_TILES + WAVES_PER_BLOCK - 1) / WAVES_PER_BLOCK;

    init_deg_kernel<<<(N_NODES + TPB - 1) / TPB, TPB, 0, stream>>>(deg);
    prep_edges_kernel<<<(N_EDGES + TPB - 1) / TPB, TPB, 0, stream>>>(
        src64, dst64, src32, dst32, deg);
    dinv_kernel<<<(N_NODES + TPB - 1) / TPB, TPB, 0, stream>>>(deg, dinv);

    gemm1_kernel<<<gemm_grid, TPB, 0, stream>>>(x, W1, h1);
    init_agg1_kernel<<<(N_NODES * HIDDEN + TPB - 1) / TPB, TPB, 0, stream>>>(
        h1, b1, dinv, agg1);
    scatter1_kernel<<<(N_EDGES * HIDDEN + TPB - 1) / TPB, TPB, 0, stream>>>(
        h1, src32, dst32, dinv, agg1);

    gemm2_kernel<<<gemm_grid, TPB, 0, stream>>>(agg1, W2, z2);
    init_out_kernel<<<(N_NODES * NUM_CLASSES + TPB - 1) / TPB, TPB, 0, stream>>>(
        z2, b2, dinv, out);
    scatter2_kernel<<<(N_EDGES * NUM_CLASSES + TPB - 1) / TPB, TPB, 0, stream>>>(
        z2, src32, dst32, dinv, out);

    log_softmax_kernel<<<(N_NODES + TPB - 1) / TPB, TPB, 0, stream>>>(out);
}